// FilterLayer_13632226197635
// MI455X (gfx1250) — compile-verified
//
#include <hip/hip_runtime.h>
#include <hip/hip_bf16.h>

#define NBINS 257
#define NF    80
#define NBP   82            // nfilt + 2 breakpoints
#define KCHUNK 64           // K rows per staged chunk
#define NCHUNK 4            // 4 chunks cover K = 0..255
#define MROWS (32 * 4096)   // 131072
// swizzled-B sizes (float2 units)
#define FRAG_PER_CHUNK (16 * 5 * 32)          // ksteps * ntiles * lanes = 2560
#define FRAG_TOTAL     (NCHUNK * FRAG_PER_CHUNK) // 10240

typedef float v2f __attribute__((ext_vector_type(2)));
typedef float v8f __attribute__((ext_vector_type(8)));

// ---------------------------------------------------------------------------
// Kernel 0: stable rank-sort of the 82 breakpoints (matches jnp.sort).
// ---------------------------------------------------------------------------
__global__ void sort_bp(const float* __restrict__ bp, float* __restrict__ bsort) {
    __shared__ float s[NBP];
    int i = threadIdx.x;
    if (i < NBP) s[i] = bp[i];
    __syncthreads();
    if (i < NBP) {
        float key = s[i];
        int r = 0;
        for (int j = 0; j < NBP; ++j) {
            float v = s[j];
            r += (v < key) || (v == key && j < i);
        }
        bsort[r] = key;
    }
}

// ---------------------------------------------------------------------------
// fbank^T value at (bin k, filter f). Row f==NF-1 is the reference's zero row.
// ---------------------------------------------------------------------------
__device__ __forceinline__ float fbank_val(const float* bs, int k, int f) {
    if (f >= NF - 1) return 0.0f;
    float bj  = bs[f];
    float bj1 = bs[f + 1];
    float bj2 = bs[f + 2];
    float lj  = floorf(bj);
    float lj1 = floorf(bj1);
    float lj2 = floorf(bj2);
    float kk  = (float)k;
    float dr  = (bj1 - bj) * (bj1 - bj);
    float df  = (bj2 - bj1) * (bj2 - bj1);
    dr = (dr == 0.0f) ? 1.0f : dr;
    df = (df == 0.0f) ? 1.0f : df;
    float val = 0.0f;
    if (kk >= lj  && kk < lj1) val += (kk - bj) / dr;
    if (kk >= lj1 && kk < lj2) val += (bj2 - kk) / df;
    return val;
}

// ---------------------------------------------------------------------------
// Kernel 1: build B pre-swizzled into per-lane WMMA fragment order:
//   fbS[kc][ks][n][lane] = { B[k + 2*half][n*16 + l16],
//                            B[k + 2*half + 1][n*16 + l16] }
// with k = kc*64 + ks*4, half = lane>>4, l16 = lane&15, B[k][f] = fbank^T.
// Also writes the K tail row fbTail[f] = fbank^T[256][f].
// ---------------------------------------------------------------------------
__global__ void build_fbS(const float* __restrict__ bsort,
                          v2f* __restrict__ fbS,
                          float* __restrict__ fbTail) {
    __shared__ float bs[NBP];
    if (threadIdx.x < NBP) bs[threadIdx.x] = bsort[threadIdx.x];
    __syncthreads();

    int idx = blockIdx.x * blockDim.x + threadIdx.x;
    if (idx < FRAG_TOTAL) {
        int lane = idx & 31;
        int n    = (idx >> 5) % 5;
        int ks   = (idx / 160) & 15;
        int kc   = idx / FRAG_PER_CHUNK;
        int half = lane >> 4;
        int l16  = lane & 15;
        int krow = kc * KCHUNK + ks * 4 + 2 * half;
        int col  = n * 16 + l16;
        v2f b;
        b.x = fbank_val(bs, krow, col);
        b.y = fbank_val(bs, krow + 1, col);
        fbS[idx] = b;
    } else if (idx < FRAG_TOTAL + NF) {
        int f = idx - FRAG_TOTAL;
        fbTail[f] = fbank_val(bs, 256, f);
    }
}

// ---------------------------------------------------------------------------
// Kernel 2: out[M,80] = x[M,257] @ fbank^T[257,80] via V_WMMA_F32_16X16X4_F32.
// 8 waves/block; each wave: 16-row M tile x all 5 N tiles. K loop = 4
// LDS-staged swizzled chunks of 64 (B frag = one ds_load_b64) + scalar tail.
// ---------------------------------------------------------------------------
__global__ __launch_bounds__(256) void fbank_gemm(const float* __restrict__ x,
                                                  const v2f* __restrict__ fbS,
                                                  const float* __restrict__ fbTail,
                                                  float* __restrict__ out) {
    __shared__ __align__(16) v2f sB[FRAG_PER_CHUNK];   // 20480 B

    const int lane = threadIdx.x & 31;
    const int wave = threadIdx.x >> 5;
    const int half = lane >> 4;     // lane half selects K pair
    const int l16  = lane & 15;     // M index for A, N index for B/C/D
    const long M0  = ((long)blockIdx.x * 8 + wave) * 16;

    v8f c[5] = {};                  // 5 N-tiles of 16x16 f32 accumulators

    // A fragment base: lanes 0-15 and 16-31 both cover M = M0 + (lane&15);
    // lane half picks K=(0,1) vs K=(2,3).
    const float* xrow = x + (M0 + l16) * NBINS + half * 2;

    for (int kc = 0; kc < NCHUNK; ++kc) {
        // Stage pre-swizzled 64-K chunk of B into LDS (contiguous float4 copy).
        const float4* src = (const float4*)(fbS + kc * FRAG_PER_CHUNK);
        float4* dst = (float4*)sB;
        for (int i = threadIdx.x; i < FRAG_PER_CHUNK / 2; i += 256)
            dst[i] = src[i];
        __syncthreads();

        const float* xk = xrow + kc * KCHUNK;
        #pragma unroll
        for (int ks = 0; ks < KCHUNK / 4; ++ks) {
            v2f a;
            a.x = xk[ks * 4];
            a.y = xk[ks * 4 + 1];
            const v2f* bfrag = sB + ks * 160 + lane;   // [ks][n][lane]
            #pragma unroll
            for (int n = 0; n < 5; ++n) {
                v2f b = bfrag[n * 32];                 // single ds_load_b64
                // (neg_a, A, neg_b, B, c_mod, C, reuse_a, reuse_b)
                c[n] = __builtin_amdgcn_wmma_f32_16x16x4_f32(
                    false, a, false, b, (short)0, c[n], false, false);
            }
        }
        __syncthreads();
    }

    // K tail: k == 256 (scalar FMA into accumulator fragments).
    {
        float fT[5];
        #pragma unroll
        for (int n = 0; n < 5; ++n)
            fT[n] = fbTail[n * 16 + l16];
        #pragma unroll
        for (int v = 0; v < 8; ++v) {
            long row = M0 + v + half * 8;   // C layout: M = vgpr + 8*lane_half
            float xv = x[row * NBINS + 256];
            #pragma unroll
            for (int n = 0; n < 5; ++n)
                c[n][v] += xv * fT[n];
        }
    }

    // Store D; fuse filtered[:,:,0] = x[:,:,0]  (N==0 <=> n==0 && l16==0).
    #pragma unroll
    for (int v = 0; v < 8; ++v) {
        long row = M0 + v + half * 8;
        #pragma unroll
        for (int n = 0; n < 5; ++n) {
            float val = c[n][v];
            if (n == 0 && l16 == 0) val = x[row * NBINS];
            out[row * NF + n * 16 + l16] = val;
        }
    }
}

// ---------------------------------------------------------------------------
extern "C" void kernel_launch(void* const* d_in, const int* in_sizes, int n_in,
                              void* d_out, int out_size, void* d_ws, size_t ws_size,
                              hipStream_t stream) {
    const float* x  = (const float*)d_in[0];
    const float* bp = (const float*)d_in[1];
    float* out = (float*)d_out;

    float* ws     = (float*)d_ws;
    float* bsort  = ws;                         // 82 floats
    v2f*   fbS    = (v2f*)(ws + 128);           // 10240 float2 (16B-aligned)
    float* fbTail = ws + 128 + 2 * FRAG_TOTAL;  // 80 floats

    sort_bp<<<1, 128, 0, stream>>>(bp, bsort);

    const int nbuild = FRAG_TOTAL + NF;
    build_fbS<<<(nbuild + 255) / 256, 256, 0, stream>>>(bsort, fbS, fbTail);

    const int blocks = MROWS / 128;   // 8 waves * 16 rows each = 128 rows/block
    fbank_gemm<<<blocks, 256, 0, stream>>>(x, fbS, fbTail, out);
}